// GraphGCNResNorm_20968030339262
// MI455X (gfx1250) — compile-verified
//
#include <hip/hip_runtime.h>

#define DCH 256
#define ALPHA 0.1f
#define EPSN 1e-6f

typedef __attribute__((ext_vector_type(2))) float v2f;
typedef __attribute__((ext_vector_type(8))) float v8f;

#define WMMA_F32(a, b, c) \
  __builtin_amdgcn_wmma_f32_16x16x4_f32(false, (a), false, (b), (short)0, (c), false, false)

// ---------------- init: zero agg, deg=1 (self loop), zero colsum ----------------
__global__ __launch_bounds__(256) void init_kernel(float* __restrict__ deg,
                                                   float* __restrict__ agg,
                                                   float* __restrict__ colsum,
                                                   int n, long long aggElems) {
  long long i = (long long)blockIdx.x * blockDim.x + threadIdx.x;
  long long stride = (long long)gridDim.x * blockDim.x;
  for (long long t = i; t < aggElems; t += stride) agg[t] = 0.0f;
  for (long long t = i; t < n; t += stride) deg[t] = 1.0f;   // self loop
  if (i < DCH) colsum[i] = 0.0f;
}

// ---------------- degree accumulation over edges ----------------
__global__ __launch_bounds__(256) void deg_kernel(const long long* __restrict__ dst,
                                                  float* __restrict__ deg, int E) {
  int i = blockIdx.x * blockDim.x + threadIdx.x;
  int stride = gridDim.x * blockDim.x;
  for (int e = i; e < E; e += stride) atomicAdd(&deg[(int)dst[e]], 1.0f);
}

__global__ __launch_bounds__(256) void dinv_kernel(const float* __restrict__ deg,
                                                   float* __restrict__ dinv, int n) {
  int i = blockIdx.x * blockDim.x + threadIdx.x;
  if (i < n) {
    float d = deg[i];
    dinv[i] = d > 0.0f ? rsqrtf(d) : 0.0f;
  }
}

// ---------------- C[n x 256] = A[n x 256] @ W^T[256 x 256], fp32 WMMA ----------------
// One wave computes a 16x64 output slab (4 col tiles): A-fragment reused 4x,
// 4 independent accumulator chains so v_wmma issues back-to-back without RAW stalls.
__global__ __launch_bounds__(256) void gemm_xwT_wmma(const float* __restrict__ A,
                                                     const float* __restrict__ W,
                                                     float* __restrict__ C,
                                                     int rowTiles) {
  int lane = threadIdx.x & 31;
  int wave = threadIdx.x >> 5;
  int wt = blockIdx.x * 8 + wave;      // wave-slab id
  if (wt >= rowTiles * 4) return;      // 4 slabs of 64 cols across D=256
  int rowT = wt >> 2;
  int colG = wt & 3;
  int rsel  = lane & 15;               // row-within-tile (A) / col-within-tile (B)
  int khalf = (lane >> 4) * 2;         // lanes 16..31 hold K+2,K+3

  const float* ap = A + (long long)(rowT * 16 + rsel) * DCH + khalf;
  const float* bp = W + (long long)(colG * 64 + rsel) * DCH + khalf;

  v8f c0 = {}, c1 = {}, c2 = {}, c3 = {};
  for (int k0 = 0; k0 < DCH; k0 += 4) {
    v2f a  = *(const v2f*)(ap + k0);
    v2f b0 = *(const v2f*)(bp + k0);
    v2f b1 = *(const v2f*)(bp + 16 * DCH + k0);
    v2f b2 = *(const v2f*)(bp + 32 * DCH + k0);
    v2f b3 = *(const v2f*)(bp + 48 * DCH + k0);
    c0 = WMMA_F32(a, b0, c0);
    c1 = WMMA_F32(a, b1, c1);
    c2 = WMMA_F32(a, b2, c2);
    c3 = WMMA_F32(a, b3, c3);
  }
  long long baseRow = (long long)rowT * 16 + (lane >> 4) * 8;
  int col0 = colG * 64 + rsel;
#pragma unroll
  for (int v = 0; v < 8; ++v) {
    float* crow = C + (baseRow + v) * DCH + col0;
    crow[0]  = c0[v];
    crow[16] = c1[v];
    crow[32] = c2[v];
    crow[48] = c3[v];
  }
}

// ---------------- fused residual dual GEMM: out = (1-a)*ax@W1^T + a*x0@W2^T ----------------
__global__ __launch_bounds__(256) void dual_gemm_wmma(const float* __restrict__ AX,
                                                      const float* __restrict__ W1,
                                                      const float* __restrict__ X0,
                                                      const float* __restrict__ W2,
                                                      float* __restrict__ C,
                                                      int rowTiles) {
  int lane = threadIdx.x & 31;
  int wave = threadIdx.x >> 5;
  int wt = blockIdx.x * 8 + wave;
  if (wt >= rowTiles * 4) return;
  int rowT = wt >> 2;
  int colG = wt & 3;
  int rsel  = lane & 15;
  int khalf = (lane >> 4) * 2;

  long long arow = (long long)(rowT * 16 + rsel) * DCH + khalf;
  long long brow = (long long)(colG * 64 + rsel) * DCH + khalf;

  v8f c0 = {}, c1 = {}, c2 = {}, c3 = {};
  {
    const float* ap = AX + arow;
    const float* bp = W1 + brow;
    for (int k0 = 0; k0 < DCH; k0 += 4) {          // (1-alpha) * ax @ W1^T
      v2f a  = *(const v2f*)(ap + k0) * (1.0f - ALPHA);
      v2f b0 = *(const v2f*)(bp + k0);
      v2f b1 = *(const v2f*)(bp + 16 * DCH + k0);
      v2f b2 = *(const v2f*)(bp + 32 * DCH + k0);
      v2f b3 = *(const v2f*)(bp + 48 * DCH + k0);
      c0 = WMMA_F32(a, b0, c0);
      c1 = WMMA_F32(a, b1, c1);
      c2 = WMMA_F32(a, b2, c2);
      c3 = WMMA_F32(a, b3, c3);
    }
  }
  {
    const float* ap = X0 + arow;
    const float* bp = W2 + brow;
    for (int k0 = 0; k0 < DCH; k0 += 4) {          // + alpha * x0 @ W2^T
      v2f a  = *(const v2f*)(ap + k0) * ALPHA;
      v2f b0 = *(const v2f*)(bp + k0);
      v2f b1 = *(const v2f*)(bp + 16 * DCH + k0);
      v2f b2 = *(const v2f*)(bp + 32 * DCH + k0);
      v2f b3 = *(const v2f*)(bp + 48 * DCH + k0);
      c0 = WMMA_F32(a, b0, c0);
      c1 = WMMA_F32(a, b1, c1);
      c2 = WMMA_F32(a, b2, c2);
      c3 = WMMA_F32(a, b3, c3);
    }
  }
  long long baseRow = (long long)rowT * 16 + (lane >> 4) * 8;
  int col0 = colG * 64 + rsel;
#pragma unroll
  for (int v = 0; v < 8; ++v) {
    float* crow = C + (baseRow + v) * DCH + col0;
    crow[0]  = c0[v];
    crow[16] = c1[v];
    crow[32] = c2[v];
    crow[48] = c3[v];
  }
}

// ---------------- edge scatter: agg[dst] += h[src] * dinv[src]*dinv[dst] ----------------
// One wave per edge (wave-strided); lane covers 2 float4 chunks of the 256-ch row.
__global__ __launch_bounds__(256) void scatter_kernel(const long long* __restrict__ src,
                                                      const long long* __restrict__ dst,
                                                      const float* __restrict__ h,
                                                      const float* __restrict__ dinv,
                                                      float* __restrict__ agg, int E) {
  int lane = threadIdx.x & 31;
  int gw = (blockIdx.x * blockDim.x + threadIdx.x) >> 5;
  int nw = (gridDim.x * blockDim.x) >> 5;
  for (int e = gw; e < E; e += nw) {
    int s = (int)src[e];
    int d = (int)dst[e];
    float nrm = dinv[s] * dinv[d];
    const float4* hp = (const float4*)(h + (long long)s * DCH);
    float* ap = agg + (long long)d * DCH;
#pragma unroll
    for (int q = 0; q < 2; ++q) {
      int c4 = lane + q * 32;          // 64 float4 per row
      float4 v = hp[c4];
      atomicAdd(ap + c4 * 4 + 0, v.x * nrm);
      atomicAdd(ap + c4 * 4 + 1, v.y * nrm);
      atomicAdd(ap + c4 * 4 + 2, v.z * nrm);
      atomicAdd(ap + c4 * 4 + 3, v.w * nrm);
    }
  }
}

// ---------------- fold self-loop contribution + bias into agg (= ax) ----------------
__global__ __launch_bounds__(256) void ax_finalize(const float* __restrict__ h,
                                                   const float* __restrict__ dinv,
                                                   const float* __restrict__ bias,
                                                   float* __restrict__ agg,
                                                   long long total) {
  long long i = (long long)blockIdx.x * blockDim.x + threadIdx.x;
  long long stride = (long long)gridDim.x * blockDim.x;
  for (long long t = i; t < total; t += stride) {
    int node = (int)(t >> 8);
    int ch = (int)(t & 255);
    float di = dinv[node];
    agg[t] = agg[t] + h[t] * di * di + bias[ch];
  }
}

// ---------------- per-channel sum of squares over nodes ----------------
__global__ __launch_bounds__(256) void colsq_kernel(const float* __restrict__ outp,
                                                    float* __restrict__ colsum, int n) {
  int ch = threadIdx.x;  // 256 channels
  int rowsPer = (n + gridDim.x - 1) / gridDim.x;
  int r0 = blockIdx.x * rowsPer;
  int r1 = r0 + rowsPer;
  if (r1 > n) r1 = n;
  float s = 0.0f;
  for (int r = r0; r < r1; ++r) {
    float v = outp[(long long)r * DCH + ch];
    s += v * v;
  }
  atomicAdd(&colsum[ch], s);
}

// ---------------- rms normalize + affine + relu ----------------
__global__ __launch_bounds__(256) void norm_relu_kernel(const float* __restrict__ outp,
                                                        const float* __restrict__ colsum,
                                                        const float* __restrict__ gamma,
                                                        const float* __restrict__ beta,
                                                        float* __restrict__ out,
                                                        int n, long long total) {
  long long i = (long long)blockIdx.x * blockDim.x + threadIdx.x;
  long long stride = (long long)gridDim.x * blockDim.x;
  float invN = 1.0f / (float)n;
  for (long long t = i; t < total; t += stride) {
    int ch = (int)(t & 255);
    float inv = rsqrtf(colsum[ch] * invN + EPSN);
    float v = outp[t] * inv * gamma[ch] + beta[ch];
    out[t] = v > 0.0f ? v : 0.0f;
  }
}

extern "C" void kernel_launch(void* const* d_in, const int* in_sizes, int n_in,
                              void* d_out, int out_size, void* d_ws, size_t ws_size,
                              hipStream_t stream) {
  const float* x      = (const float*)d_in[0];
  const float* x0     = (const float*)d_in[1];
  const long long* ei = (const long long*)d_in[2];
  const float* Wg     = (const float*)d_in[3];
  const float* bg     = (const float*)d_in[4];
  const float* W1     = (const float*)d_in[5];
  const float* W2     = (const float*)d_in[6];
  const float* gamma  = (const float*)d_in[7];
  const float* beta   = (const float*)d_in[8];

  int n = in_sizes[0] / DCH;
  int E = in_sizes[2] / 2;
  const long long* srcIdx = ei;
  const long long* dstIdx = ei + E;

  size_t nd = (size_t)n * DCH;
  char* wsp = (char*)d_ws;
  float* h      = (float*)wsp; wsp += nd * sizeof(float);
  float* agg    = (float*)wsp; wsp += nd * sizeof(float);
  float* deg    = (float*)wsp; wsp += (size_t)n * sizeof(float);
  float* dinv   = (float*)wsp; wsp += (size_t)n * sizeof(float);
  float* colsum = (float*)wsp; wsp += DCH * sizeof(float);
  // h is reused as out_pre after ax_finalize (h no longer needed then).

  int rowTiles = n / 16;                        // 50000/16 = 3125
  int waveSlabs = rowTiles * 4;                 // 16x64 slabs
  int gemmBlocks = (waveSlabs + 7) / 8;

  init_kernel<<<2048, 256, 0, stream>>>(deg, agg, colsum, n, (long long)nd);
  deg_kernel<<<2048, 256, 0, stream>>>(dstIdx, deg, E);
  dinv_kernel<<<(n + 255) / 256, 256, 0, stream>>>(deg, dinv, n);
  gemm_xwT_wmma<<<gemmBlocks, 256, 0, stream>>>(x, Wg, h, rowTiles);
  scatter_kernel<<<8192, 256, 0, stream>>>(srcIdx, dstIdx, h, dinv, agg, E);
  ax_finalize<<<4096, 256, 0, stream>>>(h, dinv, bg, agg, (long long)nd);
  dual_gemm_wmma<<<gemmBlocks, 256, 0, stream>>>(agg, W1, x0, W2, h, rowTiles);
  colsq_kernel<<<512, 256, 0, stream>>>(h, colsum, n);
  norm_relu_kernel<<<4096, 256, 0, stream>>>(h, colsum, gamma, beta,
                                             (float*)d_out, n, (long long)nd);

  // pass-through outputs: x0, then edge_index (raw bytes, capped to d_out tail)
  hipMemcpyAsync((float*)d_out + nd, x0, nd * sizeof(float),
                 hipMemcpyDeviceToDevice, stream);
  if ((size_t)out_size > 2 * nd) {
    size_t tail_bytes = ((size_t)out_size - 2 * nd) * sizeof(float);
    size_t ei_bytes = (size_t)E * 2 * sizeof(long long);
    size_t cb = tail_bytes < ei_bytes ? tail_bytes : ei_bytes;
    hipMemcpyAsync((float*)d_out + 2 * nd, (const void*)ei, cb,
                   hipMemcpyDeviceToDevice, stream);
  }
}